// Sampler_50285477101740
// MI455X (gfx1250) — compile-verified
//
#include <hip/hip_runtime.h>
#include <stdint.h>

// MI455X / gfx1250 sampler kernel (wave32).
// One 1024-thread workgroup (32 wave32s, the per-WGP max) per batch row;
// grid = B = 128 rows. Maximizes per-WGP memory-level parallelism since the
// grid can occupy at most 128 WGPs.
//
// Per row (V = 128000 floats = 512 KB; whole logits tensor = 65.5 MB, fits
// the 192 MB L2):
//   Pass 1 : stream logits (global_load_b128 + global_prefetch_b8), row
//            max/argmax, 2048-bin histogram of top-11 sortable-key bits
//            (ds_add atomics).                                  [HBM pass]
//   Pass 2 : mid-11-bit histogram for the selected prefix; simultaneously
//            collect elements strictly above the selected top bin (they are
//            guaranteed top-k survivors).                       [L2 pass]
//   Pass 3 : low-10-bit histogram; collect the mid-tier survivors and stash
//            all 22-bit-prefix-matching candidates in LDS.      [L2 pass]
//   LDS    : parallel suffix-scan bin selection each round (no serial
//            dependent-load chains), exact k-th-largest key, filter the
//            candidate list, bitonic-sort <=1024 (key<<32|idx) descending,
//            Hillis-Steele scan of exp(x-m) -> nucleus cutoff excl < p*Z,
//            then argmax of (x - log q) gathering only ~k noise entries
//            (~4 KB instead of 512 KB per row).
// Greedy rows (temp < eps) exit after pass 1 with argmax(logits).
//
// No GEMM exists in this op, so WMMA is intentionally absent; the gfx1250
// signals are wave32 codegen, dual-issue VALU, b128 vector loads, ds_*
// atomics, split wait counters, and global_prefetch_b8.

#define TPB 1024
#define CAP 1024       // >= max top_k (1000)
#define CANDCAP 2048   // 22-bit-prefix candidate guard (expect ~rank2 entries)
#define NB1 2048
#define NB2 2048
#define NB3 1024
#define SAMPLING_EPS 1e-5f

__device__ __forceinline__ uint32_t f2key(float f) {
    uint32_t u = __float_as_uint(f);
    return (u & 0x80000000u) ? ~u : (u | 0x80000000u);
}
__device__ __forceinline__ float key2f(uint32_t k) {
    uint32_t u = (k & 0x80000000u) ? (k ^ 0x80000000u) : ~k;
    return __uint_as_float(u);
}

// Iterate one row: x = logits[i] * invt, call f(x, i). b128 path when V%4==0.
template <typename F>
__device__ __forceinline__ void row_foreach(const float* __restrict__ p, int V,
                                            float invt, int tid, F f) {
    if ((V & 3) == 0) {
        const float4* p4 = reinterpret_cast<const float4*>(p);
        const int V4 = V >> 2;
        for (int i = tid; i < V4; i += TPB) {
            __builtin_prefetch(p4 + i + 4 * TPB, 0, 1);  // global_prefetch_b8
            float4 v = p4[i];
            int b = i << 2;
            f(v.x * invt, b);
            f(v.y * invt, b + 1);
            f(v.z * invt, b + 2);
            f(v.w * invt, b + 3);
        }
    } else {
        for (int i = tid; i < V; i += TPB) f(p[i] * invt, i);
    }
}

// Parallel radix-round bin selection. hist[0..NB) holds counts (synced by
// caller). Computes inclusive suffix sums S(b) with a Hillis-Steele scan
// (ping-pong hist<->hsum), then finds the unique b* with
//   S(b*) >= rank > S(b*+1),
// i.e. the bin containing the rank-th largest element when scanning bins
// from high to low. Writes *s_sel = b*, *s_rank = rank - S(b*+1).
// Destroys hist/hsum. Ends with __syncthreads().
template <int NB>
__device__ __forceinline__ void select_bin(uint32_t* hist, uint32_t* hsum,
                                           uint32_t rank, int tid,
                                           uint32_t* s_sel, uint32_t* s_rank) {
    if (tid == 0) { *s_sel = 0u; *s_rank = 1u; }  // fallback (never expected)
    uint32_t* a = hist;
    uint32_t* b = hsum;
    for (int off = 1; off < NB; off <<= 1) {
        for (int i = tid; i < NB; i += TPB)
            b[i] = a[i] + ((i + off < NB) ? a[i + off] : 0u);
        __syncthreads();
        uint32_t* t = a; a = b; b = t;
    }
    for (int i = tid; i < NB; i += TPB) {
        uint32_t Sb = a[i];
        uint32_t Sn = (i + 1 < NB) ? a[i + 1] : 0u;
        if (Sb >= rank && Sn < rank) { *s_sel = (uint32_t)i; *s_rank = rank - Sn; }
    }
    __syncthreads();
}

__global__ __launch_bounds__(TPB) void Sampler_50285477101740_kernel(
    const float* __restrict__ logits, const float* __restrict__ temps,
    const int* __restrict__ top_k, const float* __restrict__ top_p,
    const float* __restrict__ noise, int* __restrict__ out, int V) {
    const int row = blockIdx.x;
    const int tid = threadIdx.x;
    const float* lrow = logits + (size_t)row * (size_t)V;

    __shared__ uint32_t hist[NB1];
    __shared__ uint32_t hsum[NB1];
    __shared__ float red_v[TPB];
    __shared__ int red_i[TPB];
    __shared__ unsigned long long keys[CAP];
    __shared__ unsigned long long cand[CANDCAP];
    __shared__ float sva[CAP];
    __shared__ float svb[CAP];
    __shared__ uint32_t s_cnt;
    __shared__ uint32_t s_cand;
    __shared__ uint32_t s_sel;
    __shared__ uint32_t s_rank;

    const float temp = temps[row];
    const bool greedy = temp < SAMPLING_EPS;  // uniform per block
    const float invt = greedy ? 1.0f : (1.0f / temp);
    int k = top_k[row];
    if (k > V) k = V;
    if (k < 1) k = 1;
    const float tp = top_p[row];

    // ------------ Pass 1: max + argmax + top-11-bit histogram -----------
    for (int i = tid; i < NB1; i += TPB) hist[i] = 0u;
    __syncthreads();

    float bmax = -INFINITY;
    int bidx = 0;
    row_foreach(lrow, V, invt, tid, [&](float x, int i) {
        if (x > bmax) { bmax = x; bidx = i; }  // first-occurrence ties
        if (!greedy) atomicAdd(&hist[f2key(x) >> 21], 1u);  // ds_add
    });

    red_v[tid] = bmax;
    red_i[tid] = bidx;
    __syncthreads();
    for (int s = TPB / 2; s > 0; s >>= 1) {
        if (tid < s) {
            float ov = red_v[tid + s];
            int oi = red_i[tid + s];
            if (ov > red_v[tid] || (ov == red_v[tid] && oi < red_i[tid])) {
                red_v[tid] = ov;
                red_i[tid] = oi;
            }
        }
        __syncthreads();
    }
    const float m = red_v[0];   // row max of x (== max of kept set)
    const int amax = red_i[0];  // greedy argmax (invt==1 for greedy rows)

    if (greedy) {
        if (tid == 0) out[row] = amax;
        return;
    }

    // ------------ Round 1 selection (bits 31:21) ------------------------
    select_bin<NB1>(hist, hsum, (uint32_t)k, tid, &s_sel, &s_rank);
    const uint32_t sel1 = s_sel;
    const uint32_t rank1 = s_rank;

    // ------------ Pass 2: mid-11-bit histogram + collect top tier -------
    if (tid == 0) s_cnt = 0u;
    for (int i = tid; i < NB2; i += TPB) hist[i] = 0u;
    __syncthreads();
    row_foreach(lrow, V, invt, tid, [&](float x, int i) {
        uint32_t kk = f2key(x);
        uint32_t hi = kk >> 21;
        if (hi == sel1) {
            atomicAdd(&hist[(kk >> 10) & 0x7ffu], 1u);
        } else if (hi > sel1) {  // strictly above selected bin: in top-k
            uint32_t p = atomicAdd(&s_cnt, 1u);
            if (p < CAP)
                keys[p] = ((unsigned long long)kk << 32) | (uint32_t)i;
        }
    });
    __syncthreads();

    // ------------ Round 2 selection (bits 20:10) ------------------------
    select_bin<NB2>(hist, hsum, rank1, tid, &s_sel, &s_rank);
    const uint32_t sel2 = s_sel;
    const uint32_t rank2 = s_rank;
    const uint32_t pref22 = (sel1 << 21) | (sel2 << 10);

    // ------------ Pass 3: low-10-bit hist + collect mid tier + stash ----
    if (tid == 0) s_cand = 0u;
    for (int i = tid; i < NB3; i += TPB) hist[i] = 0u;
    __syncthreads();
    row_foreach(lrow, V, invt, tid, [&](float x, int i) {
        uint32_t kk = f2key(x);
        if ((kk >> 10) == (pref22 >> 10)) {  // exact 22-bit prefix match
            atomicAdd(&hist[kk & 0x3ffu], 1u);
            uint32_t p = atomicAdd(&s_cand, 1u);
            if (p < CANDCAP)
                cand[p] = ((unsigned long long)kk << 32) | (uint32_t)i;
        } else if ((kk >> 21) == sel1 && ((kk >> 10) & 0x7ffu) > sel2) {
            uint32_t p = atomicAdd(&s_cnt, 1u);  // mid tier: in top-k
            if (p < CAP)
                keys[p] = ((unsigned long long)kk << 32) | (uint32_t)i;
        }
    });
    __syncthreads();

    // ------------ Round 3 selection (bits 9:0) -> exact threshold -------
    const uint32_t ncand = (s_cand < (uint32_t)CANDCAP) ? s_cand : (uint32_t)CANDCAP;
    select_bin<NB3>(hist, hsum, rank2, tid, &s_sel, &s_rank);
    const uint32_t thr_key = pref22 | s_sel;  // exact bits of k-th largest x

    // Filter stashed candidates against the exact threshold (no global pass)
    for (int i = tid; i < (int)ncand; i += TPB) {
        uint32_t kk = (uint32_t)(cand[i] >> 32);
        if (kk >= thr_key) {
            uint32_t p = atomicAdd(&s_cnt, 1u);
            if (p < CAP) keys[p] = cand[i];
        }
    }
    __syncthreads();
    const int n = (int)(s_cnt < (uint32_t)CAP ? s_cnt : (uint32_t)CAP);
    for (int i = n + tid; i < CAP; i += TPB) keys[i] = 0ull;  // pad: sorts last
    __syncthreads();

    // ------------ Bitonic sort (descending) of 1024 u64 keys ------------
    // Total order on (value-key, index) -> deterministic result regardless of
    // the nondeterministic atomic collection order.
    for (int kk2 = 2; kk2 <= CAP; kk2 <<= 1) {
        for (int j = kk2 >> 1; j > 0; j >>= 1) {
            for (int i = tid; i < CAP; i += TPB) {
                int ixj = i ^ j;
                if (ixj > i) {
                    bool dir = ((i & kk2) != 0);  // flipped -> descending
                    unsigned long long a = keys[i];
                    unsigned long long b = keys[ixj];
                    if ((a > b) == dir) { keys[i] = b; keys[ixj] = a; }
                }
            }
            __syncthreads();
        }
    }

    // ------------ Inclusive scan of w = exp(x - m) ----------------------
    for (int i = tid; i < CAP; i += TPB) {
        float w = 0.0f;
        if (i < n) {
            float x = key2f((uint32_t)(keys[i] >> 32));
            w = expf(x - m);
        }
        sva[i] = w;
    }
    __syncthreads();
    float* a = sva;
    float* b = svb;
    for (int off = 1; off < CAP; off <<= 1) {
        for (int i = tid; i < CAP; i += TPB)
            b[i] = a[i] + ((i >= off) ? a[i - off] : 0.0f);
        __syncthreads();
        float* t = a; a = b; b = t;
    }
    const float Z = a[CAP - 1];
    const float P = tp * Z;  // keep while exclusive cumulative mass < P

    // ------------ Nucleus filter + exponential-race argmax --------------
    float best = -INFINITY;
    int bi = 0x7fffffff;
    for (int i = tid; i < n; i += TPB) {
        float x = key2f((uint32_t)(keys[i] >> 32));
        float w = expf(x - m);
        float excl = a[i] - w;  // mass strictly above this element
        if (excl < P) {         // kept by top-p (i==0 always kept)
            int idx = (int)(uint32_t)(keys[i] & 0xffffffffull);
            float q = noise[(size_t)row * (size_t)V + (size_t)idx];
            float s = x - logf(q);  // argmax(probs/q) == argmax(x - log q)
            if (s > best || (s == best && idx < bi)) { best = s; bi = idx; }
        }
    }
    red_v[tid] = best;
    red_i[tid] = bi;
    __syncthreads();
    for (int s2 = TPB / 2; s2 > 0; s2 >>= 1) {
        if (tid < s2) {
            float ov = red_v[tid + s2];
            int oi = red_i[tid + s2];
            if (ov > red_v[tid] || (ov == red_v[tid] && oi < red_i[tid])) {
                red_v[tid] = ov;
                red_i[tid] = oi;
            }
        }
        __syncthreads();
    }
    if (tid == 0) out[row] = red_i[0];
}

extern "C" void kernel_launch(void* const* d_in, const int* in_sizes, int n_in,
                              void* d_out, int out_size, void* d_ws, size_t ws_size,
                              hipStream_t stream) {
    (void)n_in; (void)d_ws; (void)ws_size; (void)out_size;
    const float* logits = (const float*)d_in[0];
    const float* temps  = (const float*)d_in[1];
    const int*   topk   = (const int*)d_in[2];   // int inputs -> const int*
    const float* topp   = (const float*)d_in[3];
    const float* noise  = (const float*)d_in[4];
    const int B = in_sizes[1];                   // temperatures: [B]
    const int V = in_sizes[0] / B;               // logits: [B, V]
    int* out = (int*)d_out;                      // reference output is int32

    hipLaunchKernelGGL(Sampler_50285477101740_kernel, dim3(B), dim3(TPB), 0,
                       stream, logits, temps, topk, topp, noise, out, V);
}